// GridPointer_41223096107413
// MI455X (gfx1250) — compile-verified
//
#include <hip/hip_runtime.h>

// Problem constants (from reference)
#define B_DIM 16
#define M_DIM 512
#define H_DIM 12
#define D_DIM 64
#define NEG_INF_F 1000000000000.0f

typedef __attribute__((ext_vector_type(2))) float v2f;
typedef __attribute__((ext_vector_type(8))) float v8f;

// -------------------------------------------------------------------------
// Kernel 1: RoPE rotation. x is (b, m, h, 128) = [q(64) | k(64)] per head.
// One thread per (b, m, h, pair j), j in [0,32). Writes rotated Q and K into
// workspace laid out as [bh][row][64] f32 row-major -- the layout that makes
// both WMMA A-tiles and B-tiles a contiguous float2 load per lane.
// -------------------------------------------------------------------------
__global__ __launch_bounds__(256) void rope_kernel(
    const float* __restrict__ x,
    float* __restrict__ qws,
    float* __restrict__ kws)
{
    int idx = blockIdx.x * blockDim.x + threadIdx.x;
    int j = idx & 31;            // rotation pair index (d/2)
    int t = idx >> 5;
    int h = t % H_DIM;
    t /= H_DIM;
    int m = t & (M_DIM - 1);
    int b = t >> 9;

    const float* xp = x + ((((size_t)b * M_DIM + m) * H_DIM + h) * 2 * D_DIM);
    v2f q = *(const v2f*)(xp + 2 * j);
    v2f k = *(const v2f*)(xp + D_DIM + 2 * j);

    // inv_freq[j] = 10000^(-j/32) = exp(j * -ln(10000)/32)
    float ang = (float)m * __expf((float)j * -0.28782313662425576f);
    float c = __cosf(ang);
    float s = __sinf(ang);

    v2f qo, ko;
    qo.x = q.x * c - q.y * s;  qo.y = q.y * c + q.x * s;
    ko.x = k.x * c - k.y * s;  ko.y = k.y * c + k.x * s;

    size_t off = (((size_t)(b * H_DIM + h) * M_DIM + m) * D_DIM) + 2 * j;
    *(v2f*)(qws + off) = qo;
    *(v2f*)(kws + off) = ko;
}

// -------------------------------------------------------------------------
// Kernel 2: per (b,h) GEMM  logits = Q (512x64) * K^T (64x512)  via
// V_WMMA_F32_16X16X4_F32, plus mask epilogue.
//
// grid.x = B*H*32 (one block per (bh, m-tile)); 256 threads = 8 waves;
// wave w owns output columns [w*64, w*64+64) of a 16-row strip (4 tiles).
//
// ISA f32 A-tile (16x4) layout: lanes 0-15 hold row M=lane, VGPR0/1 = K
// {k0,k0+1}; lanes 16-31 hold K {k0+2,k0+3}  -> one float2 per lane at
// Q[mBase + lane%16][k0 + 2*(lane/16)]. B (4x16, from K^T) is symmetric:
// one float2 per lane at K[nBase + lane%16][k0 + 2*(lane/16)].
// -------------------------------------------------------------------------
__global__ __launch_bounds__(256) void qk_wmma_kernel(
    const float* __restrict__ qws,
    const float* __restrict__ kws,
    const float* __restrict__ amask,
    float* __restrict__ out)
{
    int bid  = blockIdx.x;
    int mt   = bid & 31;          // m-tile index 0..31
    int bh   = bid >> 5;          // 0..191
    int b    = bh / H_DIM;

    int lane = threadIdx.x & 31;
    int wave = threadIdx.x >> 5;

    int r  = lane & 15;           // row/col within tile
    int kh = lane >> 4;           // 0 or 1 -> +2 in K

    int mBase  = mt * 16;
    int nBase0 = wave * 64;

    const float* Aq = qws + (size_t)bh * M_DIM * D_DIM;
    const float* Bk = kws + (size_t)bh * M_DIM * D_DIM;

    const float* Arow = Aq + (size_t)(mBase + r) * D_DIM + 2 * kh;

    v8f acc0 = {}, acc1 = {}, acc2 = {}, acc3 = {};

    #pragma unroll
    for (int k0 = 0; k0 < D_DIM; k0 += 4) {
        v2f a = *(const v2f*)(Arow + k0);
        v2f b0 = *(const v2f*)(Bk + (size_t)(nBase0 +  0 + r) * D_DIM + 2 * kh + k0);
        v2f b1 = *(const v2f*)(Bk + (size_t)(nBase0 + 16 + r) * D_DIM + 2 * kh + k0);
        v2f b2 = *(const v2f*)(Bk + (size_t)(nBase0 + 32 + r) * D_DIM + 2 * kh + k0);
        v2f b3 = *(const v2f*)(Bk + (size_t)(nBase0 + 48 + r) * D_DIM + 2 * kh + k0);
        acc0 = __builtin_amdgcn_wmma_f32_16x16x4_f32(false, a, false, b0, (short)0, acc0, false, false);
        acc1 = __builtin_amdgcn_wmma_f32_16x16x4_f32(false, a, false, b1, (short)0, acc1, false, false);
        acc2 = __builtin_amdgcn_wmma_f32_16x16x4_f32(false, a, false, b2, (short)0, acc2, false, false);
        acc3 = __builtin_amdgcn_wmma_f32_16x16x4_f32(false, a, false, b3, (short)0, acc3, false, false);
    }

    // Epilogue. C/D layout: VGPR v, lane -> M = mBase + v + 8*kh, N = nBase + r.
    const float* pm = amask + (size_t)b * M_DIM;
    const float inv_sqrt_d = 0.125f;

    v8f accs[4] = {acc0, acc1, acc2, acc3};
    #pragma unroll
    for (int t = 0; t < 4; ++t) {
        int n = nBase0 + t * 16 + r;
        float pad = pm[n];
        float padTerm = (1.0f - pad) * NEG_INF_F;
        #pragma unroll
        for (int v = 0; v < 8; ++v) {
            int m = mBase + v + 8 * kh;
            float val = accs[t][v] * pad - padTerm;
            if (n < m) val -= NEG_INF_F;     // tril(k=-1) causal penalty
            out[((size_t)bh * M_DIM + m) * M_DIM + n] = val * inv_sqrt_d;
        }
    }
}

// -------------------------------------------------------------------------
// d_in[0] = x              (float, 16*512*1536)
// d_in[1] = attention_mask (float, 16*512)
// d_in[2] = token_type_ids (int, unused by reference math)
// d_out   = logits         (float, 16*12*512*512)
// d_ws    : needs 2 * 16*12*512*64 * 4 = 50,331,648 bytes
// -------------------------------------------------------------------------
extern "C" void kernel_launch(void* const* d_in, const int* in_sizes, int n_in,
                              void* d_out, int out_size, void* d_ws, size_t ws_size,
                              hipStream_t stream)
{
    const float* x     = (const float*)d_in[0];
    const float* amask = (const float*)d_in[1];
    float* out = (float*)d_out;

    float* qws = (float*)d_ws;
    float* kws = qws + (size_t)B_DIM * H_DIM * M_DIM * D_DIM;

    // RoPE: B*M*H*32 threads
    int rope_threads = B_DIM * M_DIM * H_DIM * 32;
    rope_kernel<<<rope_threads / 256, 256, 0, stream>>>(x, qws, kws);

    // GEMM: one block per (bh, m-tile)
    int gemm_blocks = B_DIM * H_DIM * 32;
    qk_wmma_kernel<<<gemm_blocks, 256, 0, stream>>>(qws, kws, amask, out);
}